// DecodeYoloV1_39127152067180
// MI455X (gfx1250) — compile-verified
//
#include <hip/hip_runtime.h>
#include <stdint.h>

// ---------------------------------------------------------------------------
// YOLO decode + greedy NMS for MI455X (gfx1250).
// Only image 0 of the batch contributes to the output (reference uses dec[0]),
// so we decode just x[0] (4.6 MB) and run NMS in a single persistent
// 1024-thread workgroup with LDS-resident scores.
// CDNA5-specific data path: Tensor Data Mover (tensor_load_to_lds) stages the
// 54 KB confidence array into LDS, synchronized with s_wait_tensorcnt.
// ---------------------------------------------------------------------------

#define NBOX   13520      // 5 anchors * 52 * 52
#define NPAD   13568      // padded SoA stride in workspace (floats)
#define HW     2704       // 52*52
#define GRIDW  52
#define NCLS   80
#define NCH    85         // 5 + NUM_CLASSES
#define MAXDET 300
#define CONF_T 0.25f
#define IOU_T  0.5f

typedef unsigned int u32x4 __attribute__((ext_vector_type(4)));
typedef unsigned int u32x8 __attribute__((ext_vector_type(8)));

__device__ __forceinline__ float sigm(float v) { return 1.0f / (1.0f + __expf(-v)); }

// ---------------------------------------------------------------------------
// Kernel 1: decode x[0] -> SoA workspace [cls | conf | cx | cy | w | h]
// Thread t handles box t = a*2704 + h*52 + w. Per-channel reads are coalesced
// across adjacent threads (stride HW between channels within a thread).
// ---------------------------------------------------------------------------
__global__ void yolo_decode_kernel(const float* __restrict__ x,
                                   const float* __restrict__ anchors,
                                   const int*   __restrict__ insz,
                                   float* __restrict__ ws) {
  int t = blockIdx.x * blockDim.x + threadIdx.x;
  if (t >= NBOX) return;
  int a  = t / HW;
  int hw = t - a * HW;
  int gy = hw / GRIDW;
  int gx = hw - gy * GRIDW;

  const float* base = x + (size_t)(a * NCH) * HW + hw;
  float p0 = base[0];
  float p1 = base[(size_t)1 * HW];
  float p2 = base[(size_t)2 * HW];
  float p3 = base[(size_t)3 * HW];
  float p4 = base[(size_t)4 * HW];

  // argmax over class logits (sigmoid is monotone -> same argmax, same ties)
  float best = -__builtin_inff();
  int   bi   = 0;
#pragma unroll 5
  for (int c = 0; c < NCLS; ++c) {
    float v = base[(size_t)(5 + c) * HW];
    if (v > best) { best = v; bi = c; }
  }

  float stride = (float)insz[0] / (float)GRIDW;   // 416/52 = 8
  float conf = sigm(p4);
  float bx = (sigm(p0) + (float)gx) * stride;
  float by = (sigm(p1) + (float)gy) * stride;
  float bw = __expf(p2) * anchors[2 * a + 0] * stride;
  float bh = __expf(p3) * anchors[2 * a + 1] * stride;

  ws[0 * NPAD + t] = (float)bi;
  ws[1 * NPAD + t] = conf;
  ws[2 * NPAD + t] = bx;
  ws[3 * NPAD + t] = by;
  ws[4 * NPAD + t] = bw;
  ws[5 * NPAD + t] = bh;
}

// ---------------------------------------------------------------------------
// Kernel 2: greedy NMS, single workgroup (32 wave32s), 300 serial rounds.
// LDS: score[NBOX] (54 KB, filled by TDM) + reduction arrays (~8 KB).
// ---------------------------------------------------------------------------
__launch_bounds__(1024, 1)
__global__ void yolo_nms_kernel(const float* __restrict__ ws,
                                float* __restrict__ out) {
  __shared__ float score[NBOX];
  __shared__ float redV[1024];
  __shared__ int   redI[1024];
  __shared__ float ch[6];
  __shared__ int   done;

  const float* cls  = ws + 0 * NPAD;
  const float* conf = ws + 1 * NPAD;
  const float* cxA  = ws + 2 * NPAD;
  const float* cyA  = ws + 3 * NPAD;
  const float* bwA  = ws + 4 * NPAD;
  const float* bhA  = ws + 5 * NPAD;
  const int tid = threadIdx.x;

  // ---- TDM: DMA conf[0..NBOX) (54 KB contiguous) into LDS score[] ----------
  // 2-D descriptor: tensor_dim0 = tile_dim0 = NBOX, dim1 = tile_dim1 = 1,
  // data_size = 4 B, D# type = 2 ("image"). Only wave 0 issues the DMA
  // (tensor ops are per-wave, EXEC-ignored); tracked by TENSORcnt.
  if (tid < 32) {
    uint64_t ga  = (uint64_t)(uintptr_t)(const void*)conf;
    uint32_t lof = (uint32_t)(uintptr_t)(void*)&score[0]; // ISA: LDS addr = flat[31:0]
    u32x4 g0;
    g0.x = 1u;                                            // count=1, user mode
    g0.y = lof;                                           // lds_addr
    g0.z = (uint32_t)ga;                                  // global_addr[31:0]
    g0.w = (uint32_t)((ga >> 32) & 0x1FFFFFFull)          // global_addr[56:32]
         | 0x80000000u;                                   // type=2
    u32x8 g1;
    g1[0] = (2u << 16);                                   // data_size = 4 bytes
    g1[1] = ((uint32_t)NBOX & 0xFFFFu) << 16;             // tensor_dim0[15:0]
    g1[2] = (((uint32_t)NBOX >> 16) & 0xFFFFu)            // tensor_dim0[31:16]
          | (1u << 16);                                   // tensor_dim1[15:0] = 1
    g1[3] = ((uint32_t)NBOX & 0xFFFFu) << 16;             // tile_dim0 = NBOX
    g1[4] = 1u;                                           // tile_dim1 = 1
    g1[5] = (uint32_t)NBOX;                               // tensor_dim0_stride lo32
    g1[6] = 0u;
    g1[7] = 0u;
    asm volatile("tensor_load_to_lds %0, %1" :: "s"(g0), "s"(g1) : "memory");
#if __has_builtin(__builtin_amdgcn_s_wait_tensorcnt)
    __builtin_amdgcn_s_wait_tensorcnt(0);
#else
    asm volatile("s_wait_tensorcnt 0x0" ::: "memory");
#endif
  }
  __syncthreads();

  // active mask folded into score: inactive = -inf
  for (int i = tid; i < NBOX; i += 1024)
    score[i] = (score[i] > CONF_T) ? score[i] : -__builtin_inff();
  __syncthreads();

  for (int it = 0; it < MAXDET; ++it) {
    // ---- argmax (first-occurrence tie-break, matching jnp.argmax) ----------
    float bv = -__builtin_inff();
    int   bidx = 0;
    for (int i = tid; i < NBOX; i += 1024) {
      float v = score[i];
      if (v > bv) { bv = v; bidx = i; }          // strict > keeps lowest index
    }
    redV[tid] = bv; redI[tid] = bidx;
    __syncthreads();
    for (int s = 512; s > 0; s >>= 1) {
      if (tid < s) {
        float v2 = redV[tid + s]; int i2 = redI[tid + s];
        if (v2 > redV[tid] || (v2 == redV[tid] && i2 < redI[tid])) {
          redV[tid] = v2; redI[tid] = i2;
        }
      }
      __syncthreads();
    }

    if (tid == 0) {
      float v = redV[0];
      if (!(v > -__builtin_inff())) {            // no active box left
        done = 1;
      } else {
        done = 0;
        int i = redI[0];
        ch[0] = cls[i]; ch[1] = conf[i];
        ch[2] = cxA[i]; ch[3] = cyA[i];
        ch[4] = bwA[i]; ch[5] = bhA[i];
        out[it * 6 + 0] = ch[0];
        out[it * 6 + 1] = ch[1];
        out[it * 6 + 2] = ch[2];
        out[it * 6 + 3] = ch[3];
        out[it * 6 + 4] = ch[4];
        out[it * 6 + 5] = ch[5];
        out[6 * MAXDET + it] = 1.0f;             // valid mask
        score[i] = -__builtin_inff();            // .at[i].set(False)
      }
    }
    __syncthreads();

    if (done) {  // uniform branch: zero-pad remaining dets + validity
      for (int j = it * 6 + tid; j < 6 * MAXDET; j += 1024) out[j] = 0.0f;
      for (int j = 6 * MAXDET + it + tid; j < 7 * MAXDET; j += 1024) out[j] = 0.0f;
      break;
    }

    // ---- suppression sweep (repo's buggy (c-w)/2 corners + [0,1] clamp) ----
    float ccls = ch[0];
    float cxm = (ch[2] - ch[4]) * 0.5f, cym = (ch[3] - ch[5]) * 0.5f;
    float cxM = (ch[2] + ch[4]) * 0.5f, cyM = (ch[3] + ch[5]) * 0.5f;
    float ca  = fabsf((cxM - cxm) * (cyM - cym));
    for (int i = tid; i < NBOX; i += 1024) {
      if (!(score[i] > -__builtin_inff())) continue;   // already inactive
      if (cls[i] != ccls) continue;                    // class-gated NMS
      float xm = (cxA[i] - bwA[i]) * 0.5f, ym = (cyA[i] - bhA[i]) * 0.5f;
      float xM = (cxA[i] + bwA[i]) * 0.5f, yM = (cyA[i] + bhA[i]) * 0.5f;
      float iw = fminf(cxM, xM) - fmaxf(cxm, xm);
      float ih = fminf(cyM, yM) - fmaxf(cym, ym);
      iw = fminf(fmaxf(iw, 0.0f), 1.0f);               // repo clamps to [0,1]
      ih = fminf(fmaxf(ih, 0.0f), 1.0f);
      float inter = iw * ih;
      float a2 = fabsf((xM - xm) * (yM - ym));
      float iou = inter / (ca + a2 - inter + 1e-6f);
      if (iou >= IOU_T) score[i] = -__builtin_inff();
    }
    __syncthreads();
  }
}

// ---------------------------------------------------------------------------
extern "C" void kernel_launch(void* const* d_in, const int* in_sizes, int n_in,
                              void* d_out, int out_size, void* d_ws, size_t ws_size,
                              hipStream_t stream) {
  (void)in_sizes; (void)n_in; (void)out_size; (void)ws_size;
  const float* x       = (const float*)d_in[0];   // (64, 425, 52, 52) f32
  const float* anchors = (const float*)d_in[1];   // (5, 2) f32
  const int*   insz    = (const int*)d_in[2];     // scalar 416
  float* out = (float*)d_out;                     // 300*6 dets + 300 valid
  float* ws  = (float*)d_ws;                      // 6 * NPAD floats SoA

  yolo_decode_kernel<<<(NBOX + 255) / 256, 256, 0, stream>>>(x, anchors, insz, ws);
  yolo_nms_kernel<<<1, 1024, 0, stream>>>(ws, out);
}